// DeepModuleNet_14963666059384
// MI455X (gfx1250) — compile-verified
//
#include <hip/hip_runtime.h>
#include <hip/hip_bf16.h>
#include <math.h>

#define NNODES 50000
#define H1HEADS 8
#define CH1 32
#define HC1 256          // HEADS*HID
#define OUTC 64
#define MLPH 128
#define KCL 16
#define NEG_SLOPE 0.2f
#define SEPS 1e-16f

typedef __attribute__((ext_vector_type(2))) float v2f;
typedef __attribute__((ext_vector_type(8))) float v8f;

// ============ f32 WMMA GEMM: C[M,Nc] = act(A[M,K] @ B[K,Nc] + bias) ============
// One wave32 per 16x16 output tile, K stepped by 4 through V_WMMA_F32_16X16X4_F32.
// A-frag: lane holds row M=lane%16, K = (lane/16)*2 + {0,1}  (ISA 7.12.2, 32-bit A 16x4)
// B-frag: lane holds col N=lane%16, K = (lane/16)*2 + {0,1}  (column-major striping)
// C/D   : VGPR v -> M = v + 8*(lane/16), N = lane%16
__global__ __launch_bounds__(32)
void gemm_wmma_f32(const float* __restrict__ A, const float* __restrict__ B,
                   const float* __restrict__ bias, float* __restrict__ C,
                   int K, int Nc, int act) {
  const int tm   = blockIdx.x * 16;
  const int tn   = blockIdx.y * 16;
  const int lane = threadIdx.x & 31;
  const int hh   = lane >> 4;
  const int l16  = lane & 15;
  v8f acc = {};
  const float* arow = A + (size_t)(tm + l16) * K;
  #pragma unroll 4
  for (int k0 = 0; k0 < K; k0 += 4) {
    const int ka = k0 + hh * 2;
    v2f a, b;
    a.x = arow[ka + 0];
    a.y = arow[ka + 1];
    b.x = B[(size_t)(ka + 0) * Nc + tn + l16];
    b.y = B[(size_t)(ka + 1) * Nc + tn + l16];
    acc = __builtin_amdgcn_wmma_f32_16x16x4_f32(false, a, false, b,
                                                (short)0, acc, false, false);
  }
  #pragma unroll
  for (int v = 0; v < 8; ++v) {
    const int row = tm + hh * 8 + v;
    const int col = tn + l16;
    float val = acc[v];
    if (bias) val += bias[col];
    if (act == 1) val = fmaxf(val, 0.0f);
    C[(size_t)row * Nc + col] = val;
  }
}

// ============ helpers ============
__device__ __forceinline__ void edge_sd(const int* __restrict__ ei, int E, int e,
                                        int& s, int& d) {
  if (e < E) { s = ei[e]; d = ei[E + e]; }
  else       { s = e - E; d = e - E; }      // appended self-loops
}

__device__ __forceinline__ float lrelu(float v) {
  return v > 0.0f ? v : NEG_SLOPE * v;
}

__device__ float atomicMaxF(float* addr, float val) {
  unsigned int* ua = (unsigned int*)addr;
  unsigned int old = *ua;
  while (__uint_as_float(old) < val) {
    unsigned int assumed = old;
    old = atomicCAS(ua, assumed, __float_as_uint(val));
    if (old == assumed) break;
  }
  return __uint_as_float(old);
}

__global__ void fill_k(float* __restrict__ p, float v, int n) {
  int i = blockIdx.x * blockDim.x + threadIdx.x;
  if (i < n) p[i] = v;
}

// a_src[n,h] = sum_c H[n,h,c]*att_s[h,c]; a_dst likewise
__global__ void attn_dot_k(const float* __restrict__ Hm, const float* __restrict__ att_s,
                           const float* __restrict__ att_d, float* __restrict__ asrc,
                           float* __restrict__ adst, int H, int CH) {
  int idx = blockIdx.x * blockDim.x + threadIdx.x;     // n*H + h
  if (idx >= NNODES * H) return;
  int n = idx / H, h = idx - n * H;
  const float* hp = Hm + ((size_t)n * H + h) * CH;
  const float* as = att_s + h * CH;
  const float* ad = att_d + h * CH;
  float ss = 0.f, dd = 0.f;
  for (int c = 0; c < CH; ++c) { float hv = hp[c]; ss += hv * as[c]; dd += hv * ad[c]; }
  asrc[idx] = ss;
  adst[idx] = dd;
}

// pass 1: segment max of leaky_relu(a_src[src]+a_dst[dst]) over dst
__global__ void edge_max_k(const int* __restrict__ ei, int E, int Etot,
                           const float* __restrict__ asrc, const float* __restrict__ adst,
                           float* __restrict__ mbuf, int H) {
  int idx = blockIdx.x * blockDim.x + threadIdx.x;     // e*H + h
  if (idx >= Etot * H) return;
  int e = idx / H, h = idx - e * H;
  int s, d; edge_sd(ei, E, e, s, d);
  float v = lrelu(asrc[s * H + h] + adst[d * H + h]);
  atomicMaxF(&mbuf[d * H + h], v);
}

// pass 2: ex = exp(e - m[dst]); den[dst] += ex
__global__ void edge_expsum_k(const int* __restrict__ ei, int E, int Etot,
                              const float* __restrict__ asrc, const float* __restrict__ adst,
                              const float* __restrict__ mbuf, float* __restrict__ exbuf,
                              float* __restrict__ den, int H) {
  int idx = blockIdx.x * blockDim.x + threadIdx.x;
  if (idx >= Etot * H) return;
  int e = idx / H, h = idx - e * H;
  int s, d; edge_sd(ei, E, e, s, d);
  float v  = lrelu(asrc[s * H + h] + adst[d * H + h]);
  float ex = expf(v - mbuf[d * H + h]);
  exbuf[idx] = ex;
  atomicAdd(&den[d * H + h], ex);
}

// pass 3: agg[dst,h,:] += (ex/(den+eps)) * H[src,h,:]
__global__ void edge_agg_k(const int* __restrict__ ei, int E, int Etot,
                           const float* __restrict__ exbuf, const float* __restrict__ den,
                           const float* __restrict__ Hm, float* __restrict__ agg,
                           int H, int CH) {
  int idx = blockIdx.x * blockDim.x + threadIdx.x;     // e*H + h
  if (idx >= Etot * H) return;
  int e = idx / H, h = idx - e * H;
  int s, d; edge_sd(ei, E, e, s, d);
  float alpha = exbuf[idx] / (den[d * H + h] + SEPS);
  const float* hs = Hm  + ((size_t)s * H + h) * CH;
  float*       ao = agg + ((size_t)d * H + h) * CH;
  for (int c = 0; c < CH; ++c) atomicAdd(ao + c, alpha * hs[c]);
}

// in-place: p = elu(p + bias[col])
__global__ void elu_bias_k(float* __restrict__ p, const float* __restrict__ b,
                           int total, int C) {
  int i = blockIdx.x * blockDim.x + threadIdx.x;
  if (i >= total) return;
  float v = p[i] + b[i % C];
  p[i] = v > 0.0f ? v : (expf(v) - 1.0f);
}

// o = a + bias[col]
__global__ void add_bias_k(const float* __restrict__ a, const float* __restrict__ b,
                           float* __restrict__ o, int total, int C) {
  int i = blockIdx.x * blockDim.x + threadIdx.x;
  if (i >= total) return;
  o[i] = a[i] + b[i % C];
}

// logits = t[n,:] @ Wm2 + bm2 ; softmax over K=16; Wm2 staged in LDS
__global__ __launch_bounds__(128)
void mlp2_softmax_k(const float* __restrict__ t, const float* __restrict__ Wm2,
                    const float* __restrict__ bm2, float* __restrict__ s_out) {
  __shared__ float w[MLPH * KCL];
  __shared__ float bb[KCL];
  for (int i = threadIdx.x; i < MLPH * KCL; i += blockDim.x) w[i] = Wm2[i];
  if (threadIdx.x < KCL) bb[threadIdx.x] = bm2[threadIdx.x];
  __syncthreads();
  int n = blockIdx.x * blockDim.x + threadIdx.x;
  if (n >= NNODES) return;
  float logits[KCL];
  #pragma unroll
  for (int k = 0; k < KCL; ++k) logits[k] = bb[k];
  const float* tr = t + (size_t)n * MLPH;
  for (int j = 0; j < MLPH; ++j) {
    float tv = tr[j];
    #pragma unroll
    for (int k = 0; k < KCL; ++k) logits[k] += tv * w[j * KCL + k];
  }
  float mx = logits[0];
  #pragma unroll
  for (int k = 1; k < KCL; ++k) mx = fmaxf(mx, logits[k]);
  float sum = 0.f;
  #pragma unroll
  for (int k = 0; k < KCL; ++k) { logits[k] = expf(logits[k] - mx); sum += logits[k]; }
  float inv = 1.0f / sum;
  #pragma unroll
  for (int k = 0; k < KCL; ++k) s_out[(size_t)n * KCL + k] = logits[k] * inv;
}

// ================================ host ================================
static inline int cdiv(int a, int b) { return (a + b - 1) / b; }

extern "C" void kernel_launch(void* const* d_in, const int* in_sizes, int n_in,
                              void* d_out, int out_size, void* d_ws, size_t ws_size,
                              hipStream_t stream) {
  const float* x        = (const float*)d_in[0];
  const int*   ei       = (const int*)  d_in[1];
  const float* W1       = (const float*)d_in[2];
  const float* att_src1 = (const float*)d_in[3];
  const float* att_dst1 = (const float*)d_in[4];
  const float* b1       = (const float*)d_in[5];
  const float* W2       = (const float*)d_in[6];
  const float* att_src2 = (const float*)d_in[7];
  const float* att_dst2 = (const float*)d_in[8];
  const float* b2       = (const float*)d_in[9];
  const float* Wm1      = (const float*)d_in[10];
  const float* bm1      = (const float*)d_in[11];
  const float* Wm2      = (const float*)d_in[12];
  const float* bm2      = (const float*)d_in[13];

  const int N    = NNODES;
  const int E    = in_sizes[1] / 2;      // 400000
  const int Etot = E + N;                // +self loops

  // workspace layout (floats)
  float* ws   = (float*)d_ws;
  float* H1   = ws;                    // N*256  (later reused as MLP hidden N*128)
  float* AG1  = H1  + (size_t)N * HC1; // N*256  (becomes h after in-place elu+bias)
  float* H2   = AG1 + (size_t)N * HC1; // N*64
  float* AG2  = H2  + (size_t)N * OUTC;// N*64
  float* ASRC = AG2 + (size_t)N * OUTC;// N*8
  float* ADST = ASRC + (size_t)N * H1HEADS;
  float* MB   = ADST + (size_t)N * H1HEADS;
  float* DEN  = MB   + (size_t)N * H1HEADS;
  float* EX   = DEN  + (size_t)N * H1HEADS; // Etot*8 (reused with H=1 for layer 2)

  float* out_s   = (float*)d_out;          // N*16
  float* out_emb = out_s + (size_t)N * KCL; // N*64

  const int TB = 256;

  // ---------------- GAT layer 1 ----------------
  gemm_wmma_f32<<<dim3(N / 16, HC1 / 16), 32, 0, stream>>>(x, W1, nullptr, H1, 128, HC1, 0);
  attn_dot_k<<<cdiv(N * H1HEADS, TB), TB, 0, stream>>>(H1, att_src1, att_dst1, ASRC, ADST,
                                                       H1HEADS, CH1);
  fill_k<<<cdiv(N * H1HEADS, TB), TB, 0, stream>>>(MB, -INFINITY, N * H1HEADS);
  fill_k<<<cdiv(N * H1HEADS, TB), TB, 0, stream>>>(DEN, 0.0f, N * H1HEADS);
  fill_k<<<cdiv(N * HC1, TB), TB, 0, stream>>>(AG1, 0.0f, N * HC1);
  edge_max_k   <<<cdiv(Etot * H1HEADS, TB), TB, 0, stream>>>(ei, E, Etot, ASRC, ADST, MB, H1HEADS);
  edge_expsum_k<<<cdiv(Etot * H1HEADS, TB), TB, 0, stream>>>(ei, E, Etot, ASRC, ADST, MB, EX, DEN, H1HEADS);
  edge_agg_k   <<<cdiv(Etot * H1HEADS, TB), TB, 0, stream>>>(ei, E, Etot, EX, DEN, H1, AG1, H1HEADS, CH1);
  elu_bias_k<<<cdiv(N * HC1, TB), TB, 0, stream>>>(AG1, b1, N * HC1, HC1);   // AG1 -> h

  // ---------------- GAT layer 2 (1 head, no concat) ----------------
  gemm_wmma_f32<<<dim3(N / 16, OUTC / 16), 32, 0, stream>>>(AG1, W2, nullptr, H2, HC1, OUTC, 0);
  attn_dot_k<<<cdiv(N, TB), TB, 0, stream>>>(H2, att_src2, att_dst2, ASRC, ADST, 1, OUTC);
  fill_k<<<cdiv(N, TB), TB, 0, stream>>>(MB, -INFINITY, N);
  fill_k<<<cdiv(N, TB), TB, 0, stream>>>(DEN, 0.0f, N);
  fill_k<<<cdiv(N * OUTC, TB), TB, 0, stream>>>(AG2, 0.0f, N * OUTC);
  edge_max_k   <<<cdiv(Etot, TB), TB, 0, stream>>>(ei, E, Etot, ASRC, ADST, MB, 1);
  edge_expsum_k<<<cdiv(Etot, TB), TB, 0, stream>>>(ei, E, Etot, ASRC, ADST, MB, EX, DEN, 1);
  edge_agg_k   <<<cdiv(Etot, TB), TB, 0, stream>>>(ei, E, Etot, EX, DEN, H2, AG2, 1, OUTC);
  add_bias_k<<<cdiv(N * OUTC, TB), TB, 0, stream>>>(AG2, b2, out_emb, N * OUTC, OUTC);

  // ---------------- MLP head + softmax ----------------
  float* T = H1;  // reuse: N*128
  gemm_wmma_f32<<<dim3(N / 16, MLPH / 16), 32, 0, stream>>>(out_emb, Wm1, bm1, T, OUTC, MLPH, 1);
  mlp2_softmax_k<<<cdiv(N, 128), 128, 0, stream>>>(T, Wm2, bm2, out_s);
}